// SparseGrid_1829656068493
// MI455X (gfx1250) — compile-verified
//
#include <hip/hip_runtime.h>

#if defined(__has_builtin)
#  if __has_builtin(__builtin_amdgcn_wmma_f32_16x16x4_f32)
#    define HAVE_WMMA_F32_16X16X4 1
#  endif
#endif
#ifndef HAVE_WMMA_F32_16X16X4
#  define HAVE_WMMA_F32_16X16X4 0
#endif

typedef float v2f __attribute__((ext_vector_type(2)));
typedef float v8f __attribute__((ext_vector_type(8)));

namespace {
constexpr int   kReso        = 128;
constexpr float kStep        = 0.5f;
constexpr int   kRays        = 1024;
constexpr int   kWave        = 32;    // gfx1250 is wave32-only
constexpr int   kWavesPerRay = 4;     // 4 chunks of 32 steps in flight per ray
constexpr int   kChunks      = 14;    // ceil(448 / 32)
constexpr int   kSuper       = (kChunks + kWavesPerRay - 1) / kWavesPerRay;  // 4

__device__ __forceinline__ float wave_incl_scan(float v, int lane) {
#pragma unroll
  for (int off = 1; off < kWave; off <<= 1) {
    float n = __shfl_up(v, off, kWave);
    if (lane >= off) v += n;
  }
  return v;
}

__device__ __forceinline__ float wave_sum(float v) {
#pragma unroll
  for (int off = kWave / 2; off > 0; off >>= 1)
    v += __shfl_xor(v, off, kWave);
  return v;
}
}  // namespace

__global__ __launch_bounds__(kWave * kWavesPerRay) void plenoxel_march(
    const float* __restrict__ origins, const float* __restrict__ dirs,
    const float* __restrict__ density, const float* __restrict__ shdat,
    const int* __restrict__ links, float* __restrict__ out) {
  const int lane = threadIdx.x & (kWave - 1);
  const int w    = threadIdx.x >> 5;        // wave id within the ray (0..3)
  const int ray  = blockIdx.x;              // one ray per workgroup

  __shared__ float sums[2][kWavesPerRay];   // ping-pong chunk totals
  __shared__ float partial[kWavesPerRay][3];

  // ---- per-ray setup (uniform across block; compiler scalarizes to SALU) ----
  float ox = origins[3 * ray + 0], oy = origins[3 * ray + 1], oz = origins[3 * ray + 2];
  float dx = dirs[3 * ray + 0],    dy = dirs[3 * ray + 1],    dz = dirs[3 * ray + 2];
  float dn = sqrtf(dx * dx + dy * dy + dz * dz);
  float vx = dx / dn, vy = dy / dn, vz = dz / dn;

  // world -> grid (radius=1, center=0 => offset=0.5, scaling=0.5, gsz=128)
  float gox = (ox * 0.5f + 0.5f) * 128.0f - 0.5f;
  float goy = (oy * 0.5f + 0.5f) * 128.0f - 0.5f;
  float goz = (oz * 0.5f + 0.5f) * 128.0f - 0.5f;
  float g0x = vx * 64.0f, g0y = vy * 64.0f, g0z = vz * 64.0f;
  float glen = sqrtf(g0x * g0x + g0y * g0y + g0z * g0z);
  float ds   = 1.0f / glen;                               // delta_scale
  float gdx = g0x * ds, gdy = g0y * ds, gdz = g0z * ds;   // unit in grid space

  // SH degree-2 basis (svox2 convention)
  const float C1 = 0.4886025119029199f;
  float shm[9];
  shm[0] = 0.28209479177387814f;
  shm[1] = -C1 * vy;
  shm[2] =  C1 * vz;
  shm[3] = -C1 * vx;
  shm[4] =  1.0925484305920792f * vx * vy;
  shm[5] = -1.0925484305920792f * vy * vz;
  shm[6] =  0.31539156525252005f * (2.0f * vz * vz - vx * vx - vy * vy);
  shm[7] = -1.0925484305920792f * vx * vz;
  shm[8] =  0.5462742152960396f * (vx * vx - vy * vy);

  // ray / AABB intersection in grid space (matches jnp.where(gdirs==0,...))
  float tmin = -1e9f, tmax = 1e9f;
  {
    float go[3] = {gox, goy, goz};
    float gd[3] = {gdx, gdy, gdz};
#pragma unroll
    for (int a = 0; a < 3; ++a) {
      if (gd[a] != 0.0f) {
        float inv = 1.0f / gd[a];
        float t1 = (-0.5f - go[a]) * inv;
        float t2 = (127.5f - go[a]) * inv;
        tmin = fmaxf(tmin, fminf(t1, t2));
        tmax = fminf(tmax, fmaxf(t1, t2));
      }
    }
  }
  tmin = fmaxf(tmin, 0.0f);

  float carry = 0.0f;   // log-transmittance entering this super-chunk
                        // (every thread computes it identically)
  float accR = 0.0f, accG = 0.0f, accB = 0.0f;

  for (int s = 0; s < kSuper; ++s) {
    // uniform across the block (same ray): exact early exit past the AABB
    float tbase_s = tmin + kStep * (float)(kWave * kWavesPerRay * s);
    if (tbase_s > tmax) break;

    const int chunk = s * kWavesPerRay + w;
    float ts    = tmin + kStep * (float)(kWave * chunk + lane);
    bool  valid = (ts <= tmax);

    // CDNA5 prefetch (global_prefetch_b8): warm next super-chunk's link cells
    float tsn = ts + kStep * (float)(kWave * kWavesPerRay);
    if (tsn <= tmax) {
      float qx = fminf(fmaxf(gox + tsn * gdx, 0.0f), 127.0f);
      float qy = fminf(fmaxf(goy + tsn * gdy, 0.0f), 127.0f);
      float qz = fminf(fmaxf(goz + tsn * gdz, 0.0f), 127.0f);
      int nx = (int)qx; nx = nx > 126 ? 126 : nx;
      int ny = (int)qy; ny = ny > 126 ? 126 : ny;
      int nz = (int)qz; nz = nz > 126 ? 126 : nz;
      __builtin_prefetch(links + (nx * kReso + ny) * kReso + nz, 0, 0);
    }

    float la = 0.0f;                 // log attenuation for this sample
    float r0 = 0.0f, r1 = 0.0f, r2 = 0.0f;
    if (valid) {
      float px = fminf(fmaxf(gox + ts * gdx, 0.0f), 127.0f);
      float py = fminf(fmaxf(goy + ts * gdy, 0.0f), 127.0f);
      float pz = fminf(fmaxf(goz + ts * gdz, 0.0f), 127.0f);
      int lx = (int)px; lx = lx > 126 ? 126 : lx;   // floor (px >= 0)
      int ly = (int)py; ly = ly > 126 ? 126 : ly;
      int lz = (int)pz; lz = lz > 126 ? 126 : lz;
      float fx = px - (float)lx, fy = py - (float)ly, fz = pz - (float)lz;
      float wx[2] = {1.0f - fx, fx};
      float wy[2] = {1.0f - fy, fy};
      float wz[2] = {1.0f - fz, fz};
      const int* lb = links + (lx * kReso + ly) * kReso + lz;

      // gather all 8 link ids first -> one load clause
      int lk[8];
#pragma unroll
      for (int c = 0; c < 8; ++c)
        lk[c] = lb[((c >> 2) & 1) * (kReso * kReso) + ((c >> 1) & 1) * kReso + (c & 1)];

      float sigma = 0.0f;
#pragma unroll
      for (int c = 0; c < 8; ++c) {
        int cx = (c >> 2) & 1, cy = (c >> 1) & 1, cz = c & 1;
        // branch-free: invalid links -> index 0 with zero weight
        float wv  = (lk[c] >= 0) ? (wx[cx] * wy[cy] * wz[cz]) : 0.0f;
        int   idx = (lk[c] >= 0) ? lk[c] : 0;
        sigma += wv * density[idx];
        const float* sp = shdat + (long long)idx * 27;
        float a0 = 0.0f, a1 = 0.0f, a2 = 0.0f;
#pragma unroll
        for (int k = 0; k < 9; ++k) {
          a0 += shm[k] * sp[k];
          a1 += shm[k] * sp[9 + k];
          a2 += shm[k] * sp[18 + k];
        }
        r0 += wv * a0; r1 += wv * a1; r2 += wv * a2;
      }
      la = -kStep * fmaxf(sigma, 0.0f) * ds;
    }

    // intra-wave inclusive scan of log-attenuation
    float incl  = wave_incl_scan(la, lane);
    float total = __shfl(incl, kWave - 1, kWave);
    if (lane == 0) sums[s & 1][w] = total;
    __syncthreads();   // single barrier per super-chunk (ping-pong buffer)

    float t0 = sums[s & 1][0], t1 = sums[s & 1][1];
    float t2 = sums[s & 1][2], t3 = sums[s & 1][3];
    float offset = carry;                 // cross-chunk exclusive prefix
    if (w > 0) offset += t0;
    if (w > 1) offset += t1;
    if (w > 2) offset += t2;

    float cuml = offset + incl;           // cumulative incl. this sample
    if (valid) {
      float weight = expf(cuml - la) * (1.0f - expf(la));
      accR += weight * fmaxf(r0 + 0.5f, 0.0f);
      accG += weight * fmaxf(r1 + 0.5f, 0.0f);
      accB += weight * fmaxf(r2 + 0.5f, 0.0f);
    }
    carry += t0 + t1 + t2 + t3;           // identical in every thread
  }

  // ---- block reduction: 3 channels over 4 waves ----
  accR = wave_sum(accR);
  accG = wave_sum(accG);
  accB = wave_sum(accB);
  if (lane == 0) {
    partial[w][0] = accR; partial[w][1] = accG; partial[w][2] = accB;
  }
  __syncthreads();

  float bg = expf(carry);                 // * background_brightness (1.0)

#if HAVE_WMMA_F32_16X16X4
  // Cross-wave reduction via V_WMMA_F32_16X16X4_F32: D = A(16x4) * B(4x16).
  // A[m][k] = partial[k][m] (m = channel 0..2, k = wave 0..3), B = all-ones
  // (all-ones is layout-invariant), C = 0  =>  D[m][n] = sum_k partial[k][m].
  // A striping (ISA 7.12.2): lanes 0-15 VGPR0=K0, VGPR1=K1; lanes 16-31 K2,K3.
  {
    int m   = lane & 15;
    int jlo = (lane < 16) ? 0 : 2;
    float e0 = (m < 3) ? partial[jlo + 0][m] : 0.0f;
    float e1 = (m < 3) ? partial[jlo + 1][m] : 0.0f;
    v2f A; A.x = e0; A.y = e1;
    v2f B; B.x = 1.0f; B.y = 1.0f;
    v8f C = {};
    v8f D = __builtin_amdgcn_wmma_f32_16x16x4_f32(
        /*neg_a=*/false, A, /*neg_b=*/false, B,
        /*c_mod=*/(short)0, C, /*reuse_a=*/false, /*reuse_b=*/false);
    // D layout: element r, lane 0 -> D[M=r][N=0]; rows 0..2 are the channels.
    if (threadIdx.x == 0) {
      out[3 * ray + 0] = D[0] + bg;
      out[3 * ray + 1] = D[1] + bg;
      out[3 * ray + 2] = D[2] + bg;
    }
  }
#else
  if (threadIdx.x == 0) {
    float oR = bg, oG = bg, oB = bg;
#pragma unroll
    for (int j = 0; j < kWavesPerRay; ++j) {
      oR += partial[j][0]; oG += partial[j][1]; oB += partial[j][2];
    }
    out[3 * ray + 0] = oR;
    out[3 * ray + 1] = oG;
    out[3 * ray + 2] = oB;
  }
#endif
}

extern "C" void kernel_launch(void* const* d_in, const int* in_sizes, int n_in,
                              void* d_out, int out_size, void* d_ws, size_t ws_size,
                              hipStream_t stream) {
  (void)in_sizes; (void)n_in; (void)out_size; (void)d_ws; (void)ws_size;
  const float* origins = (const float*)d_in[0];
  const float* dirs    = (const float*)d_in[1];
  const float* density = (const float*)d_in[2];
  const float* shdat   = (const float*)d_in[3];
  const int*   links   = (const int*)d_in[4];
  float*       out     = (float*)d_out;

  dim3 block(kWave * kWavesPerRay);   // 128 threads = 4 wave32s = 1 ray
  dim3 grid(kRays);                   // 1024 workgroups -> 4096 waves in flight
  plenoxel_march<<<grid, block, 0, stream>>>(origins, dirs, density, shdat, links, out);
}